// VectorQuantizerST_45612552683710
// MI455X (gfx1250) — compile-verified
//
#include <hip/hip_runtime.h>
#include <hip/hip_bf16.h>

typedef __attribute__((ext_vector_type(2))) float v2f;
typedef __attribute__((ext_vector_type(8))) float v8f;

// D(16x16) = A(16x4) * B(4x16) + C, fp32, wave32.
__device__ __forceinline__ v8f wmma4(v2f a, v2f b, v8f c) {
  return __builtin_amdgcn_wmma_f32_16x16x4_f32(false, a, false, b, (short)0, c, false, false);
}

// CDNA5 async DMA: 16 bytes global -> LDS, tracked by ASYNCcnt.
__device__ __forceinline__ void async_lds_b128(unsigned lds_off, unsigned long long gaddr) {
  asm volatile("global_load_async_to_lds_b128 %0, %1, off"
               :: "v"(lds_off), "v"(gaddr) : "memory");
}
__device__ __forceinline__ void async_wait0() {
  asm volatile("s_wait_asynccnt 0x0" ::: "memory");
}

#define N_ROWS 65536
#define C_DIM  256
#define K_CODE 1024

// ---------------------------------------------------------------- init
__global__ __launch_bounds__(256) void init_kernel(unsigned* cs, unsigned* ct,
                                                   float* avgs, float* avgt, float* sumsq) {
  for (int k = threadIdx.x; k < 1024; k += 256) {
    cs[k] = 0u; ct[k] = 0u; avgs[k] = 0.f; avgt[k] = 0.f;
  }
  if (threadIdx.x == 0) sumsq[0] = 0.f;
}

// ---------------------------------------------------------------- groupnorm stats
__global__ __launch_bounds__(256) void gn_stats_kernel(const float* __restrict__ z,
                                                       float* __restrict__ mu_out,
                                                       float* __restrict__ rsig_out) {
  const int bg = blockIdx.x;
  const int b = bg >> 4, g = bg & 15;
  const float* base = z + (size_t)b * (256 * 16384) + (size_t)g * 16 * 16384;
  const int len = 16 * 16384;
  float s = 0.f, sq = 0.f;
  for (int i = threadIdx.x; i < len; i += 256) { float v = base[i]; s += v; sq += v * v; }
  __shared__ float rs[256], rq[256];
  rs[threadIdx.x] = s; rq[threadIdx.x] = sq; __syncthreads();
  for (int st = 128; st > 0; st >>= 1) {
    if (threadIdx.x < st) { rs[threadIdx.x] += rs[threadIdx.x + st]; rq[threadIdx.x] += rq[threadIdx.x + st]; }
    __syncthreads();
  }
  if (threadIdx.x == 0) {
    float mu  = rs[0] / (float)len;
    float var = rq[0] / (float)len - mu * mu;
    mu_out[bg]   = mu;
    rsig_out[bg] = rsqrtf(var + 1e-6f);
  }
}

// ---------------------------------------------------------------- transpose (b,c,t,h,w)->rows(N,C) + normalize
__global__ __launch_bounds__(256) void tn_kernel(const float* __restrict__ z,
    const float* __restrict__ mu, const float* __restrict__ rsig,
    const float* __restrict__ gamma, const float* __restrict__ beta,
    float* __restrict__ zf, float* __restrict__ hn) {
  __shared__ float tile[32 * 33];
  const int blk = blockIdx.x;
  const int ctile = blk & 7;        // channel tile of 32
  const int s = blk >> 3;           // (b*16+t)*32+h   in [0,2048)
  const int b = s >> 9;
  const int t = (s >> 5) & 15;
  const int h = s & 31;
  const int c0 = ctile * 32;
  const size_t zbase = (size_t)b * (256 * 16384) + (size_t)t * 1024 + (size_t)h * 32;
#pragma unroll
  for (int i = 0; i < 4; ++i) {
    int e = threadIdx.x + 256 * i;
    int cl = e >> 5, w = e & 31;
    tile[cl * 33 + w] = z[zbase + (size_t)(c0 + cl) * 16384 + w];
  }
  __syncthreads();
#pragma unroll
  for (int i = 0; i < 4; ++i) {
    int e = threadIdx.x + 256 * i;
    int w = e >> 5, cl = e & 31;
    int c = c0 + cl;
    float v = tile[cl * 33 + w];
    size_t n = (size_t)s * 32 + w;      // row index (b,t,h,w)
    zf[n * 256 + c] = v;
    int bg = (b << 4) | (c >> 4);
    hn[n * 256 + c] = (v - mu[bg]) * rsig[bg] * gamma[c] + beta[c];
  }
}

// ---------------------------------------------------------------- codebook row norms
__global__ __launch_bounds__(256) void codenorm_kernel(const float* __restrict__ emb,
                                                       float* __restrict__ cn) {
  int k = blockIdx.x * 256 + threadIdx.x;
  const float* r = emb + (size_t)k * 256;
  float s = 0.f;
  for (int c = 0; c < 256; ++c) { float v = r[c]; s += v * v; }
  cn[k] = s;
}

// ---------------------------------------------------------------- WMMA f32 GEMM: out = in(N,256) * W(256,256)^T + bias
// 32-row M-tile staged to LDS via async DMA; wave = 2 M-tiles x 2 N-tiles.
__global__ __launch_bounds__(256) void gemm256_kernel(const float* __restrict__ in,
    const float* __restrict__ W, const float* __restrict__ bias, float* __restrict__ out) {
  __shared__ float As[32 * 260];      // padded stride vs 64 banks (16B-aligned rows)
  const int tid = threadIdx.x;
  const int m0 = blockIdx.x * 32;
  const unsigned lds_base = (unsigned)(size_t)(&As[0]);
#pragma unroll
  for (int i = 0; i < 8; ++i) {
    int f = tid + 256 * i;            // float4 index into 32x256 tile
    int row = f >> 6, c4 = f & 63;
    async_lds_b128(lds_base + (unsigned)(row * 260 + c4 * 4) * 4u,
                   (unsigned long long)(size_t)(in + (size_t)(m0 + row) * 256 + c4 * 4));
  }
  async_wait0();
  __syncthreads();
  const int wave = tid >> 5, lane = tid & 31;
  const int half = lane >> 4, l16 = lane & 15;
  const v8f vzero = {0.f, 0.f, 0.f, 0.f, 0.f, 0.f, 0.f, 0.f};
  v8f a00 = vzero, a01 = vzero, a10 = vzero, a11 = vzero;  // [mtile][ntile]
  const int nb0 = wave * 32, nb1 = nb0 + 16;
  const float* w0  = W + (size_t)(nb0 + l16) * 256 + 2 * half;
  const float* w1  = W + (size_t)(nb1 + l16) * 256 + 2 * half;
  const float* ar0 = As + l16 * 260 + 2 * half;
  const float* ar1 = As + (16 + l16) * 260 + 2 * half;
  for (int kk = 0; kk < 64; ++kk) {
    const int k0 = kk * 4;
    v2f A0 = *(const v2f*)(ar0 + k0);
    v2f A1 = *(const v2f*)(ar1 + k0);
    v2f B0 = *(const v2f*)(w0 + k0);
    v2f B1 = *(const v2f*)(w1 + k0);
    a00 = wmma4(A0, B0, a00);
    a01 = wmma4(A0, B1, a01);
    a10 = wmma4(A1, B0, a10);
    a11 = wmma4(A1, B1, a11);
  }
  const float bv0 = bias[nb0 + l16], bv1 = bias[nb1 + l16];
#pragma unroll
  for (int r = 0; r < 8; ++r) {
    const int m = r + 8 * half;
    out[(size_t)(m0 + m) * 256 + nb0 + l16]      = a00[r] + bv0;
    out[(size_t)(m0 + m) * 256 + nb1 + l16]      = a01[r] + bv1;
    out[(size_t)(m0 + 16 + m) * 256 + nb0 + l16] = a10[r] + bv0;
    out[(size_t)(m0 + 16 + m) * 256 + nb1 + l16] = a11[r] + bv1;
  }
}

// ---------------------------------------------------------------- temporal attention per (b,h,w)
__global__ __launch_bounds__(256) void attn_kernel(const float* __restrict__ q,
    const float* __restrict__ k, const float* __restrict__ v, float* __restrict__ ho) {
  __shared__ float qs[16 * 257], ks[16 * 257], vs[16 * 257];
  __shared__ float att[16 * 17];
  const int x = blockIdx.x;           // b*1024 + h*32 + w
  const int b = x >> 10;
  const int hw = x & 1023;
  const size_t base = (size_t)b * 16384 + hw;   // row for time t: base + t*1024
#pragma unroll
  for (int i = 0; i < 16; ++i) {
    size_t ro = (base + (size_t)i * 1024) * 256 + threadIdx.x;
    qs[i * 257 + threadIdx.x] = q[ro];
    ks[i * 257 + threadIdx.x] = k[ro];
    vs[i * 257 + threadIdx.x] = v[ro];
  }
  __syncthreads();
  {
    int tq = threadIdx.x >> 4, tk = threadIdx.x & 15;
    float sdot = 0.f;
    for (int c = 0; c < 256; ++c) sdot += qs[tq * 257 + c] * ks[tk * 257 + c];
    att[tq * 17 + tk] = sdot * 0.0625f;     // * C^-0.5
  }
  __syncthreads();
  if (threadIdx.x < 16) {
    int tq = threadIdx.x;
    float m = att[tq * 17];
    for (int j = 1; j < 16; ++j) m = fmaxf(m, att[tq * 17 + j]);
    float ex[16]; float ssum = 0.f;
    for (int j = 0; j < 16; ++j) { ex[j] = expf(att[tq * 17 + j] - m); ssum += ex[j]; }
    float inv = 1.f / ssum;
    for (int j = 0; j < 16; ++j) att[tq * 17 + j] = ex[j] * inv;
  }
  __syncthreads();
  {
    int c = threadIdx.x;
    for (int tq = 0; tq < 16; ++tq) {
      float o = 0.f;
      for (int tk = 0; tk < 16; ++tk) o += vs[tk * 257 + c] * att[tq * 17 + tk];
      ho[(base + (size_t)tq * 1024) * 256 + c] = o;
    }
  }
}

// ---------------------------------------------------------------- ztf = zt + (zf[t]-zf[t-1]), in-place on zt
__global__ __launch_bounds__(256) void diffadd_kernel(const float* __restrict__ zf,
                                                      float* __restrict__ ztf) {
  size_t i = (size_t)blockIdx.x * 256 + threadIdx.x;  // over N*C
  size_t n = i >> 8;
  int t = (int)((n >> 10) & 15);
  float d = 0.f;
  if (t > 0) d = zf[i] - zf[i - (size_t)1024 * 256];
  ztf[i] += d;
}

// ---------------------------------------------------------------- fused distance GEMM + argmin + inv-dist stats
// block = 32 rows x 1024 codes; 8 waves, each 2 M-tiles x 8 N-tiles x 64 k-steps of wmma f32 16x16x4
__global__ __launch_bounds__(256) void dist_kernel(const float* __restrict__ rows,
    const float* __restrict__ emb, const float* __restrict__ codn2,
    int* __restrict__ idx_out, unsigned int* __restrict__ counts,
    float* __restrict__ avg_out) {
  __shared__ float As[32 * 260];
  __shared__ float avgp[1024];
  __shared__ float pmin[32 * 8];
  __shared__ int   parg[32 * 8];
  __shared__ float pinv[32 * 8];
  __shared__ float rn2[32];
  __shared__ float rowinv[32];
  __shared__ float red[256];

  const int tid = threadIdx.x;
  const int m0 = blockIdx.x * 32;
  const unsigned lds_base = (unsigned)(size_t)(&As[0]);
#pragma unroll
  for (int i = 0; i < 8; ++i) {
    int f = tid + 256 * i;
    int row = f >> 6, c4 = f & 63;
    async_lds_b128(lds_base + (unsigned)(row * 260 + c4 * 4) * 4u,
                   (unsigned long long)(size_t)(rows + (size_t)(m0 + row) * 256 + c4 * 4));
  }
  async_wait0();
  for (int i = tid; i < 1024; i += 256) avgp[i] = 0.f;
  __syncthreads();
  { // row squared norms: 8 threads per row, 32 elems each
    int r = tid >> 3, sg = tid & 7;
    float p = 0.f;
#pragma unroll
    for (int i = 0; i < 32; ++i) { float vv = As[r * 260 + sg * 32 + i]; p += vv * vv; }
    red[tid] = p;
  }
  __syncthreads();
  if (tid < 32) {
    float s = 0.f;
#pragma unroll
    for (int i = 0; i < 8; ++i) s += red[tid * 8 + i];
    rn2[tid] = s;
  }
  __syncthreads();

  const int wave = tid >> 5, lane = tid & 31;
  const int half = lane >> 4, l16 = lane & 15;

  const v8f vzero = {0.f, 0.f, 0.f, 0.f, 0.f, 0.f, 0.f, 0.f};
  v8f acc[2][8];
#pragma unroll
  for (int mt = 0; mt < 2; ++mt)
#pragma unroll
    for (int j = 0; j < 8; ++j) acc[mt][j] = vzero;

  const float* ar0   = As + l16 * 260 + 2 * half;
  const float* ar1   = As + (16 + l16) * 260 + 2 * half;
  const float* bbase = emb + (size_t)(wave * 128 + l16) * 256 + 2 * half;
  for (int kk = 0; kk < 64; ++kk) {
    const int k0 = kk * 4;
    v2f A0 = *(const v2f*)(ar0 + k0);
    v2f A1 = *(const v2f*)(ar1 + k0);
#pragma unroll
    for (int j = 0; j < 8; ++j) {
      v2f B = *(const v2f*)(bbase + (size_t)j * 16 * 256 + k0);
      acc[0][j] = wmma4(A0, B, acc[0][j]);
      acc[1][j] = wmma4(A1, B, acc[1][j]);
    }
  }

  // epilogue: d = |row|^2 + |code|^2 - 2 dot ; per-lane partial min/arg/invsum
  float lmin[2][8], linv[2][8]; int larg[2][8];
#pragma unroll
  for (int mt = 0; mt < 2; ++mt)
#pragma unroll
    for (int r = 0; r < 8; ++r) { lmin[mt][r] = 3.4e38f; linv[mt][r] = 0.f; larg[mt][r] = 0; }
#pragma unroll
  for (int j = 0; j < 8; ++j) {
    const int nb = wave * 128 + j * 16 + l16;
    const float cn = codn2[nb];
#pragma unroll
    for (int mt = 0; mt < 2; ++mt)
#pragma unroll
      for (int r = 0; r < 8; ++r) {
        float d = rn2[mt * 16 + r + 8 * half] + cn - 2.f * acc[mt][j][r];
        acc[mt][j][r] = d;
        linv[mt][r] += 1.f / d;
        if (d < lmin[mt][r]) { lmin[mt][r] = d; larg[mt][r] = nb; }
      }
  }
  // wave32 butterfly reduce across the 16 lanes of each half (first-index tie break)
#pragma unroll
  for (int mt = 0; mt < 2; ++mt)
#pragma unroll
    for (int r = 0; r < 8; ++r) {
      float bm = lmin[mt][r]; int ba = larg[mt][r]; float si = linv[mt][r];
#pragma unroll
      for (int mk = 1; mk <= 8; mk <<= 1) {
        float om = __shfl_xor(bm, mk, 32);
        int   oa = __shfl_xor(ba, mk, 32);
        float os = __shfl_xor(si, mk, 32);
        si += os;
        if (om < bm || (om == bm && oa < ba)) { bm = om; ba = oa; }
      }
      if (l16 == 0) {
        int m = mt * 16 + r + 8 * half;
        pmin[m * 8 + wave] = bm;
        parg[m * 8 + wave] = ba;
        pinv[m * 8 + wave] = si;
      }
    }
  __syncthreads();
  if (tid < 32) {
    float bm = 3.4e38f; int ba = 0x7fffffff; float s = 0.f;
    for (int wv = 0; wv < 8; ++wv) {
      float vm = pmin[tid * 8 + wv]; int va = parg[tid * 8 + wv];
      s += pinv[tid * 8 + wv];
      if (vm < bm || (vm == bm && va < ba)) { bm = vm; ba = va; }
    }
    rowinv[tid] = s;
    idx_out[m0 + tid] = ba;
    atomicAdd(&counts[ba], 1u);
  }
  __syncthreads();
#pragma unroll
  for (int j = 0; j < 8; ++j) {
    const int nb = wave * 128 + j * 16 + l16;
#pragma unroll
    for (int mt = 0; mt < 2; ++mt)
#pragma unroll
      for (int r = 0; r < 8; ++r) {
        int m = mt * 16 + r + 8 * half;
        atomicAdd(&avgp[nb], (1.f / acc[mt][j][r]) / rowinv[m]);
      }
  }
  __syncthreads();
  for (int i = tid; i < 1024; i += 256)
    atomicAdd(&avg_out[i], avgp[i] * (1.f / 65536.f));
}

// ---------------------------------------------------------------- z_q gather + loss sum
__global__ __launch_bounds__(256) void gather_kernel(const float* __restrict__ zf,
    const float* __restrict__ emb_s, const float* __restrict__ emb_t,
    const int* __restrict__ idx_s, const int* __restrict__ idx_t,
    float* __restrict__ out_zq, float* __restrict__ sumsq) {
  __shared__ float red[256];
  const int n = blockIdx.x;
  const int c = threadIdx.x;
  const int is = idx_s[n], it = idx_t[n];
  const float zq = emb_s[(size_t)is * 256 + c] + emb_t[(size_t)it * 256 + c];
  const float dv = zq - zf[(size_t)n * 256 + c];
  const int b = n >> 14, rem = n & 16383;
  const int t = rem >> 10, s2 = rem & 1023;
  out_zq[((size_t)(b * 256 + c) * 16 + t) * 1024 + s2] = zq;
  red[c] = dv * dv;
  __syncthreads();
  for (int st = 128; st > 0; st >>= 1) { if (c < st) red[c] += red[c + st]; __syncthreads(); }
  if (c == 0) atomicAdd(sumsq, red[0]);
}

// ---------------------------------------------------------------- index output permute (reference's (t,b,h,w) reshape)
__global__ __launch_bounds__(256) void ind_kernel(const int* __restrict__ idx_s,
    const int* __restrict__ idx_t, float* __restrict__ out_s, float* __restrict__ out_t) {
  int m = blockIdx.x * 256 + threadIdx.x;
  int j = m >> 14;            // b'
  int i = (m >> 10) & 15;     // t'
  int hw = m & 1023;
  int src = ((i * 4 + j) << 10) + hw;
  out_s[m] = (float)idx_s[src];
  out_t[m] = (float)idx_t[src];
}

// ---------------------------------------------------------------- scalars: loss, loss2(KL), perplexities
__device__ float blk_sum(float v, float* red) {
  int tid = threadIdx.x;
  red[tid] = v; __syncthreads();
  for (int st = 128; st > 0; st >>= 1) { if (tid < st) red[tid] += red[tid + st]; __syncthreads(); }
  float r = red[0]; __syncthreads();
  return r;
}
__device__ float blk_max(float v, float* red) {
  int tid = threadIdx.x;
  red[tid] = v; __syncthreads();
  for (int st = 128; st > 0; st >>= 1) { if (tid < st) red[tid] = fmaxf(red[tid], red[tid + st]); __syncthreads(); }
  float r = red[0]; __syncthreads();
  return r;
}

__global__ __launch_bounds__(256) void scalars_kernel(const float* __restrict__ sumsq,
    const unsigned* __restrict__ cs, const unsigned* __restrict__ ct,
    const float* __restrict__ avgs, const float* __restrict__ avgt,
    float* __restrict__ out) {
  __shared__ float red[256];
  const int tid = threadIdx.x;
  float hs = 0.f, ht = 0.f;
  for (int i = tid; i < 1024; i += 256) {
    float es = (float)cs[i] * (1.f / 65536.f);
    hs += es * logf(es + 1e-10f);
    float et = (float)ct[i] * (1.f / 65536.f);
    ht += et * logf(et + 1e-10f);
  }
  float Hs = blk_sum(hs, red);
  float Ht = blk_sum(ht, red);
  float ms = -3.4e38f, mt = -3.4e38f;
  for (int i = tid; i < 1024; i += 256) { ms = fmaxf(ms, avgs[i]); mt = fmaxf(mt, avgt[i]); }
  float Ms = blk_max(ms, red);
  float Mt = blk_max(mt, red);
  float es_ = 0.f, et_ = 0.f, sa = 0.f, st_ = 0.f;
  for (int i = tid; i < 1024; i += 256) {
    es_ += expf(avgs[i] - Ms); sa += avgs[i];
    et_ += expf(avgt[i] - Mt); st_ += avgt[i];
  }
  float Es = blk_sum(es_, red); float Sa = blk_sum(sa, red);
  float Et = blk_sum(et_, red); float St = blk_sum(st_, red);
  if (tid == 0) {
    const float Kf = 1024.f;
    float lse_s = Ms + logf(Es);
    float lse_t = Mt + logf(Et);
    float logtgt = -logf(Kf);
    float kls = (1.f / (Kf * Kf)) * (Kf * logtgt + Kf * lse_s - Sa);
    float klt = (1.f / (Kf * Kf)) * (Kf * logtgt + Kf * lse_t - St);
    out[(size_t)16777216] = 1.25f * sumsq[0] / (65536.f * 256.f);  // loss (stop-grad collapses)
    out[(size_t)16777217] = 0.1f * (kls + klt);                    // loss2
    out[(size_t)16908290] = expf(-Hs);                             // perp_s
    out[(size_t)16908291] = expf(-Ht);                             // perp_t
  }
}

// ---------------------------------------------------------------- launch
extern "C" void kernel_launch(void* const* d_in, const int* in_sizes, int n_in,
                              void* d_out, int out_size, void* d_ws, size_t ws_size,
                              hipStream_t stream) {
  (void)in_sizes; (void)n_in; (void)out_size; (void)ws_size;
  const float* z     = (const float*)d_in[0];
  const float* emb_s = (const float*)d_in[1];
  const float* emb_t = (const float*)d_in[2];
  const float* gamma = (const float*)d_in[3];
  const float* beta  = (const float*)d_in[4];
  const float* wq = (const float*)d_in[5];  const float* bq = (const float*)d_in[6];
  const float* wk = (const float*)d_in[7];  const float* bk = (const float*)d_in[8];
  const float* wv = (const float*)d_in[9];  const float* bv = (const float*)d_in[10];
  const float* wp = (const float*)d_in[11]; const float* bp = (const float*)d_in[12];

  const size_t NC = (size_t)N_ROWS * C_DIM;
  float* zf = (float*)d_ws;
  float* hn = zf + NC;            // groupnorm rows, later reused for ho
  float* qb = hn + NC;
  float* kb = qb + NC;            // k rows, later zt/ztf
  float* vb = kb + NC;
  float* sm = vb + NC;
  float* mu    = sm;              // 64
  float* rsig  = mu + 64;         // 64
  float* cns   = rsig + 64;       // 1024
  float* cntn  = cns + 1024;      // 1024
  float* avgs  = cntn + 1024;     // 1024
  float* avgt  = avgs + 1024;     // 1024
  float* sumsq = avgt + 1024;     // 1
  unsigned* csu = (unsigned*)(sumsq + 1);  // 1024
  unsigned* ctu = csu + 1024;              // 1024
  int* idxs = (int*)(ctu + 1024);          // 65536
  int* idxt = idxs + 65536;                // 65536

  float* out = (float*)d_out;
  float* out_zq   = out;
  float* out_inds = out + 16777218;
  float* out_indt = out + 16777218 + 65536;

  hipLaunchKernelGGL(init_kernel, dim3(1), dim3(256), 0, stream, csu, ctu, avgs, avgt, sumsq);
  hipLaunchKernelGGL(gn_stats_kernel, dim3(64), dim3(256), 0, stream, z, mu, rsig);
  hipLaunchKernelGGL(tn_kernel, dim3(16384), dim3(256), 0, stream, z, mu, rsig, gamma, beta, zf, hn);
  hipLaunchKernelGGL(codenorm_kernel, dim3(4), dim3(256), 0, stream, emb_s, cns);
  hipLaunchKernelGGL(codenorm_kernel, dim3(4), dim3(256), 0, stream, emb_t, cntn);
  hipLaunchKernelGGL(gemm256_kernel, dim3(2048), dim3(256), 0, stream, hn, wq, bq, qb);
  hipLaunchKernelGGL(gemm256_kernel, dim3(2048), dim3(256), 0, stream, hn, wk, bk, kb);
  hipLaunchKernelGGL(gemm256_kernel, dim3(2048), dim3(256), 0, stream, hn, wv, bv, vb);
  hipLaunchKernelGGL(attn_kernel, dim3(4096), dim3(256), 0, stream, qb, kb, vb, hn);   // ho -> hn
  hipLaunchKernelGGL(gemm256_kernel, dim3(2048), dim3(256), 0, stream, hn, wp, bp, kb); // zt -> kb
  hipLaunchKernelGGL(diffadd_kernel, dim3(65536), dim3(256), 0, stream, zf, kb);        // ztf in-place
  hipLaunchKernelGGL(dist_kernel, dim3(2048), dim3(256), 0, stream, zf, emb_s, cns, idxs, csu, avgs);
  hipLaunchKernelGGL(dist_kernel, dim3(2048), dim3(256), 0, stream, kb, emb_t, cntn, idxt, ctu, avgt);
  hipLaunchKernelGGL(gather_kernel, dim3(65536), dim3(256), 0, stream, zf, emb_s, emb_t, idxs, idxt, out_zq, sumsq);
  hipLaunchKernelGGL(ind_kernel, dim3(256), dim3(256), 0, stream, idxs, idxt, out_inds, out_indt);
  hipLaunchKernelGGL(scalars_kernel, dim3(1), dim3(256), 0, stream, sumsq, csu, ctu, avgs, avgt, out);
}